// Snake_WTConv2d_858993459269
// MI455X (gfx1250) — compile-verified
//
#include <hip/hip_runtime.h>
#include <hip/hip_bf16.h>
#include <stdint.h>

typedef __attribute__((ext_vector_type(2))) float v2f;
typedef __attribute__((ext_vector_type(8))) float v8f;

#define C_CH    64
#define C4_CH   256
#define KP_PER  12                    // 9 taps zero-padded to 12 (3 WMMA chunks)
#define KP_TOT  (C4_CH * KP_PER)      // 3072
#define RAW_W   18                    // 16-pixel tile + 2 halo columns
#define RAW_N   (3 * C4_CH * RAW_W)   // 13824 floats, exactly 54 per thread @256

// ---------------------------------------------------------------------------
// Prep: transposed, tap-padded A matrix:  ApadT[(ci*12+p)*16 + m]
// ---------------------------------------------------------------------------
__global__ void prep_apadT_kernel(const float* __restrict__ off_w,
                                  float* __restrict__ ApadT) {
    int i = blockIdx.x * blockDim.x + threadIdx.x;
    if (i >= KP_TOT * 16) return;
    int m  = i & 15;
    int kp = i >> 4;
    int ci = kp / KP_PER;
    int p  = kp - ci * KP_PER;
    float v = 0.0f;
    if (p < 9 && m < 6) v = off_w[(m * C4_CH + ci) * 9 + p];
    ApadT[i] = v;
}

// ---------------------------------------------------------------------------
// Base: depthwise 3x3 conv + bias, scaled -> d_out (B,C,H,W)
// ---------------------------------------------------------------------------
__global__ void base_conv_kernel(const float* __restrict__ x,
                                 const float* __restrict__ bw,
                                 const float* __restrict__ bb,
                                 const float* __restrict__ bs,
                                 float* __restrict__ out,
                                 int B, int H, int W) {
    int i = blockIdx.x * blockDim.x + threadIdx.x;
    int total = B * C_CH * H * W;
    if (i >= total) return;
    int xx = i % W;
    int yy = (i / W) % H;
    int ch = (i / (W * H)) % C_CH;
    int b  = i / (W * H * C_CH);
    const float* xp = x + ((size_t)(b * C_CH + ch) * H) * W;
    float acc = 0.0f;
#pragma unroll
    for (int ky = 0; ky < 3; ++ky) {
#pragma unroll
        for (int kx = 0; kx < 3; ++kx) {
            int y2 = yy + ky - 1, x2 = xx + kx - 1;
            int yc = min(max(y2, 0), H - 1);
            int xc = min(max(x2, 0), W - 1);
            float inb = (y2 >= 0 && y2 < H && x2 >= 0 && x2 < W) ? 1.0f : 0.0f;
            acc += bw[ch * 9 + ky * 3 + kx] * xp[(size_t)yc * W + xc] * inb;
        }
    }
    out[i] = bs[ch] * (acc + bb[ch]);
}

// ---------------------------------------------------------------------------
// Haar analysis: in (B, ch-plane stride layout) -> cx (B, 4c, h, w), h=H/2
// ---------------------------------------------------------------------------
__global__ void haar_wt_kernel(const float* __restrict__ in,
                               float* __restrict__ cx,
                               int B, int chStep, int Cmul, int H, int W) {
    int h = H >> 1, w = W >> 1;
    int i = blockIdx.x * blockDim.x + threadIdx.x;
    int total = B * C_CH * h * w;
    if (i >= total) return;
    int j  = i % w;
    int ii = (i / w) % h;
    int ch = (i / (w * h)) % C_CH;
    int b  = i / (w * h * C_CH);
    const float* p = in + ((size_t)(b * Cmul + ch * chStep) * H + 2 * ii) * W + 2 * j;
    float a = p[0], bv = p[1], c2 = p[W], d = p[W + 1];
    float ll = 0.5f * (a + bv + c2 + d);
    float hl = 0.5f * (a + bv - c2 - d);
    float lh = 0.5f * (a - bv + c2 - d);
    float hh = 0.5f * (a - bv - c2 + d);
    size_t plane = (size_t)h * w;
    size_t ob = ((size_t)(b * C4_CH + ch * 4) * h + ii) * w + j;
    cx[ob] = ll; cx[ob + plane] = hl; cx[ob + 2 * plane] = lh; cx[ob + 3 * plane] = hh;
}

// ---------------------------------------------------------------------------
// Offset conv via WMMA fp32 16x16x4 implicit GEMM, one block per 16-pixel tile.
// Interior tiles stage the 55KB tag tile with async global->LDS DMA (ASYNCcnt);
// boundary tiles use the clamp+mask VGPR path. 8 waves split K'; partial C
// reduced in LDS; 3-row means + bias -> ox,oy.
// ---------------------------------------------------------------------------
__global__ __launch_bounds__(256)
void offset_wmma_kernel(const float* __restrict__ tag,
                        const float* __restrict__ ApadT,
                        const float* __restrict__ off_b,
                        float* __restrict__ ox,
                        float* __restrict__ oy,
                        int B, int H, int W) {
    __shared__ float raw[RAW_N];               // [ky][ci][col]
    __shared__ float red[8][2][16];

    const int tid = threadIdx.x;
    const int tiles_x = W >> 4;
    const int tx = blockIdx.x % tiles_x;
    const int y  = (blockIdx.x / tiles_x) % H;
    const int bb = blockIdx.x / (tiles_x * H);
    const int x0 = tx << 4;

    const float* tagB = tag + (size_t)bb * C4_CH * H * W;

    // ---- stage raw tile ----
    const bool interior = (y > 0) && (y < H - 1) && (x0 > 0) && (x0 + 16 < W);
    if (interior) {
        // async DMA path: each lane copies one float of channel ci, row y+ky-1, column x0+col-1 of tagB to raw[s]
        const uint64_t gbase = (uint64_t)(uintptr_t)tagB;
        for (int s = tid; s < RAW_N; s += 256) {
            int ky  = s / (C4_CH * RAW_W);
            int s2  = s - ky * (C4_CH * RAW_W);
            int ci  = s2 / RAW_W;
            int col = s2 - ci * RAW_W;
            uint32_t goff = (uint32_t)((((ci * H) + (y + ky - 1)) * W +
                                        (x0 + col - 1)) * 4);
            uint32_t lds = (uint32_t)(uintptr_t)&raw[s];
            asm volatile("global_load_async_to_lds_b32 %0, %1, %2"
                         :: "v"(lds), "v"(goff), "s"(gbase) : "memory");
        }
        asm volatile("s_wait_asynccnt 0x0" ::: "memory");
    } else {
        for (int s = tid; s < RAW_N; s += 256) {
            int ky  = s / (C4_CH * RAW_W);
            int s2  = s - ky * (C4_CH * RAW_W);
            int ci  = s2 / RAW_W;
            int col = s2 - ci * RAW_W;
            int yy = y + ky - 1;
            int xx = x0 + col - 1;
            int yc = min(max(yy, 0), H - 1);
            int xc = min(max(xx, 0), W - 1);
            float inb = (yy >= 0 && yy < H && xx >= 0 && xx < W) ? 1.0f : 0.0f;
            raw[s] = tagB[((size_t)ci * H + yc) * W + xc] * inb;
        }
    }
    __syncthreads();

    // ---- WMMA over this wave's 32-channel K' slice ----
    const int lane  = tid & 31;
    const int wave  = tid >> 5;
    const int laneN = lane & 15;
    const int hi    = lane >> 4;
    const int ci0   = wave * 32;

    v8f acc = {};
    for (int cc = 0; cc < 32; ++cc) {
        const int ci = ci0 + cc;
        const float* aBase = ApadT + (size_t)ci * (KP_PER * 16) + laneN;
#pragma unroll
        for (int sub = 0; sub < 2; ++sub) {       // taps p = 0..7, always valid
            v2f a, bf;
#pragma unroll
            for (int t = 0; t < 2; ++t) {
                int p = sub * 4 + (hi << 1) + t;
                a[t] = aBase[p * 16];
                int ky = (p * 171) >> 9;          // p/3
                int kx = p - 3 * ky;
                bf[t] = raw[(ky * C4_CH + ci) * RAW_W + laneN + kx];
            }
            acc = __builtin_amdgcn_wmma_f32_16x16x4_f32(false, a, false, bf,
                                                        (short)0, acc, false, false);
        }
        {   // sub == 2: only p=8 (tap ky=2,kx=2) is real; p=9..11 are padding
            v2f a, bf;
            a[0] = aBase[(8 + (hi << 1)) * 16];
            a[1] = aBase[(9 + (hi << 1)) * 16];
            float v8 = raw[(2 * C4_CH + ci) * RAW_W + laneN + 2];
            bf[0] = hi ? 0.0f : v8;
            bf[1] = 0.0f;
            acc = __builtin_amdgcn_wmma_f32_16x16x4_f32(false, a, false, bf,
                                                        (short)0, acc, false, false);
        }
    }

    // ---- reduce partial C across 8 waves; rows 0..5 live in lanes 0..15 ----
    float sx = acc[0] + acc[1] + acc[2];
    float sy = acc[3] + acc[4] + acc[5];
    if (lane < 16) { red[wave][0][laneN] = sx; red[wave][1][laneN] = sy; }
    __syncthreads();

    if (tid < 32) {
        int which = tid >> 4;          // 0 -> ox, 1 -> oy
        int n     = tid & 15;
        float s = 0.0f;
#pragma unroll
        for (int wv = 0; wv < 8; ++wv) s += red[wv][which][n];
        const float third = 1.0f / 3.0f;
        float bm = (off_b[which * 3 + 0] + off_b[which * 3 + 1] +
                    off_b[which * 3 + 2]) * third;
        float v = s * third + bm;
        size_t o = ((size_t)bb * H + y) * W + x0 + n;
        (which ? oy : ox)[o] = v;
    }
}

// ---------------------------------------------------------------------------
// Reflect-padded bilinear grid sample
// ---------------------------------------------------------------------------
__device__ __forceinline__ float reflectf(float t, int size) {
    float m = (float)(size - 1);
    float period = 2.0f * m;
    float c = fabsf(t);
    c = fmodf(c, period);
    c = (c > m) ? (period - c) : c;
    return fminf(fmaxf(c, 0.0f), m);
}

__global__ void grid_sample_kernel(const float* __restrict__ tag,
                                   const float* __restrict__ ox,
                                   const float* __restrict__ oy,
                                   float* __restrict__ xd,
                                   int B, int H, int W) {
    int i = blockIdx.x * blockDim.x + threadIdx.x;
    int total = B * C4_CH * H * W;
    if (i >= total) return;
    int x  = i % W;
    int y  = (i / W) % H;
    int ci = (i / (W * H)) % C4_CH;
    int b  = i / (W * H * C4_CH);

    size_t po = ((size_t)b * H + y) * W + x;
    float px = reflectf((float)x + ox[po], W);
    float py = reflectf((float)y + oy[po], H);

    float x0 = floorf(px), y0 = floorf(py);
    float fx = px - x0,    fy = py - y0;
    int x0i = min(max((int)x0, 0), W - 1);
    int y0i = min(max((int)y0, 0), H - 1);
    int x1i = min(x0i + 1, W - 1);
    int y1i = min(y0i + 1, H - 1);

    const float* t = tag + (size_t)(b * C4_CH + ci) * H * W;
    float v00 = t[(size_t)y0i * W + x0i];
    float v01 = t[(size_t)y0i * W + x1i];
    float v10 = t[(size_t)y1i * W + x0i];
    float v11 = t[(size_t)y1i * W + x1i];
    xd[i] = (v00 * (1.0f - fx) + v01 * fx) * (1.0f - fy)
          + (v10 * (1.0f - fx) + v11 * fx) * fy;
}

// ---------------------------------------------------------------------------
// Fused depthwise 3x3 + ReLU + grouped 4x4 pointwise + per-channel scale
// ---------------------------------------------------------------------------
__global__ void dw_pw_kernel(const float* __restrict__ xd,
                             const float* __restrict__ dw_w,
                             const float* __restrict__ pw_w,
                             const float* __restrict__ ws,
                             float* __restrict__ out,
                             int B, int H, int W) {
    int i = blockIdx.x * blockDim.x + threadIdx.x;
    int total = B * C_CH * H * W;
    if (i >= total) return;
    int x  = i % W;
    int y  = (i / W) % H;
    int ch = (i / (W * H)) % C_CH;
    int b  = i / (W * H * C_CH);

    float dwv[4];
#pragma unroll
    for (int j = 0; j < 4; ++j) {
        int ci = ch * 4 + j;
        const float* p = xd + (size_t)(b * C4_CH + ci) * H * W;
        float acc = 0.0f;
#pragma unroll
        for (int ky = 0; ky < 3; ++ky) {
#pragma unroll
            for (int kx = 0; kx < 3; ++kx) {
                int y2 = y + ky - 1, x2 = x + kx - 1;
                int yc = min(max(y2, 0), H - 1);
                int xc = min(max(x2, 0), W - 1);
                float inb = (y2 >= 0 && y2 < H && x2 >= 0 && x2 < W) ? 1.0f : 0.0f;
                acc += dw_w[ci * 9 + ky * 3 + kx] * p[(size_t)yc * W + xc] * inb;
            }
        }
        dwv[j] = fmaxf(acc, 0.0f);
    }
#pragma unroll
    for (int band = 0; band < 4; ++band) {
        int co = ch * 4 + band;
        float s = 0.0f;
#pragma unroll
        for (int j = 0; j < 4; ++j) s += pw_w[co * 4 + j] * dwv[j];
        out[((size_t)(b * C4_CH + co) * H + y) * W + x] = ws[co] * s;
    }
}

// ---------------------------------------------------------------------------
// Haar synthesis; optional nxt added to band 0; optional accumulate into out
// ---------------------------------------------------------------------------
__global__ void iwt_kernel(const float* __restrict__ tag,
                           const float* __restrict__ nxtin,
                           float* __restrict__ out,
                           int B, int h, int w, int accumulate) {
    int H = 2 * h, W = 2 * w;
    int i = blockIdx.x * blockDim.x + threadIdx.x;
    int total = B * C_CH * H * W;
    if (i >= total) return;
    int x  = i % W;
    int y  = (i / W) % H;
    int ch = (i / (W * H)) % C_CH;
    int b  = i / (W * H * C_CH);
    int ii = y >> 1, j = x >> 1;

    size_t plane = (size_t)h * w;
    size_t base  = ((size_t)(b * C4_CH + ch * 4) * h + ii) * w + j;
    float v0 = tag[base];
    if (nxtin) v0 += nxtin[((size_t)(b * C_CH + ch) * h + ii) * w + j];
    float v1 = tag[base + plane];
    float v2 = tag[base + 2 * plane];
    float v3 = tag[base + 3 * plane];
    float sy = (y & 1) ? -1.0f : 1.0f;
    float sx = (x & 1) ? -1.0f : 1.0f;
    float r = 0.5f * (v0 + sy * v1 + sx * v2 + sy * sx * v3);
    size_t oo = ((size_t)(b * C_CH + ch) * H + y) * W + x;
    if (accumulate) out[oo] += r;
    else            out[oo] = r;
}

// ---------------------------------------------------------------------------
extern "C" void kernel_launch(void* const* d_in, const int* in_sizes, int n_in,
                              void* d_out, int out_size, void* d_ws, size_t ws_size,
                              hipStream_t stream) {
    const float* x        = (const float*)d_in[0];
    const float* base_w   = (const float*)d_in[1];
    const float* base_b   = (const float*)d_in[2];
    const float* base_s   = (const float*)d_in[3];
    const float* off_w0   = (const float*)d_in[4];
    const float* off_b0   = (const float*)d_in[5];
    const float* dw_w0    = (const float*)d_in[6];
    const float* pw_w0    = (const float*)d_in[7];
    const float* ws0      = (const float*)d_in[8];
    const float* off_w1   = (const float*)d_in[9];
    const float* off_b1   = (const float*)d_in[10];
    const float* dw_w1    = (const float*)d_in[11];
    const float* pw_w1    = (const float*)d_in[12];
    const float* ws1      = (const float*)d_in[13];
    float* out = (float*)d_out;

    const int B = 16, H0 = 128, W0 = 128;
    const int h0 = 64, w0 = 64, h1 = 32, w1 = 32;

    // workspace carve (floats)
    float* p = (float*)d_ws;
    float* cx0 = p;                p += (size_t)B * C4_CH * h0 * w0;  // tag0 / tag0_out
    float* xd0 = p;                p += (size_t)B * C4_CH * h0 * w0;
    float* cx1 = p;                p += (size_t)B * C4_CH * h1 * w1;  // tag1 / tag1_out
    float* xd1 = p;                p += (size_t)B * C4_CH * h1 * w1;  // reused as nxt1
    float* ox0 = p;                p += (size_t)B * h0 * w0;
    float* oy0 = p;                p += (size_t)B * h0 * w0;
    float* ox1 = p;                p += (size_t)B * h1 * w1;
    float* oy1 = p;                p += (size_t)B * h1 * w1;
    float* ApT0 = p;               p += (size_t)KP_TOT * 16;
    float* ApT1 = p;               p += (size_t)KP_TOT * 16;

    const int T = 256;
    auto blocks = [](long long n, int t) { return (int)((n + t - 1) / t); };

    // 1) weight prep for WMMA A matrices (transposed + tap-padded)
    prep_apadT_kernel<<<blocks((long long)KP_TOT * 16, T), T, 0, stream>>>(off_w0, ApT0);
    prep_apadT_kernel<<<blocks((long long)KP_TOT * 16, T), T, 0, stream>>>(off_w1, ApT1);

    // 2) base branch -> d_out
    base_conv_kernel<<<blocks((long long)B * C_CH * H0 * W0, T), T, 0, stream>>>(
        x, base_w, base_b, base_s, out, B, H0, W0);

    // 3) analysis level 0: x -> cx0 ; level 1: LL(cx0) -> cx1
    haar_wt_kernel<<<blocks((long long)B * C_CH * h0 * w0, T), T, 0, stream>>>(
        x, cx0, B, /*chStep=*/1, /*Cmul=*/C_CH, H0, W0);
    haar_wt_kernel<<<blocks((long long)B * C_CH * h1 * w1, T), T, 0, stream>>>(
        cx0, cx1, B, /*chStep=*/4, /*Cmul=*/C4_CH, h0, w0);

    // 4) level 0 dysnake: offsets (WMMA) -> sample -> dw+relu+pw (into cx0)
    offset_wmma_kernel<<<B * h0 * (w0 >> 4), 256, 0, stream>>>(
        cx0, ApT0, off_b0, ox0, oy0, B, h0, w0);
    grid_sample_kernel<<<blocks((long long)B * C4_CH * h0 * w0, T), T, 0, stream>>>(
        cx0, ox0, oy0, xd0, B, h0, w0);
    dw_pw_kernel<<<blocks((long long)B * C_CH * h0 * w0, T), T, 0, stream>>>(
        xd0, dw_w0, pw_w0, ws0, cx0, B, h0, w0);

    // 5) level 1 dysnake (into cx1)
    offset_wmma_kernel<<<B * h1 * (w1 >> 4), 256, 0, stream>>>(
        cx1, ApT1, off_b1, ox1, oy1, B, h1, w1);
    grid_sample_kernel<<<blocks((long long)B * C4_CH * h1 * w1, T), T, 0, stream>>>(
        cx1, ox1, oy1, xd1, B, h1, w1);
    dw_pw_kernel<<<blocks((long long)B * C_CH * h1 * w1, T), T, 0, stream>>>(
        xd1, dw_w1, pw_w1, ws1, cx1, B, h1, w1);

    // 6) synthesis: iwt level1 -> nxt1 (reuse xd1); iwt level0 + nxt1 += d_out
    iwt_kernel<<<blocks((long long)B * C_CH * h0 * w0, T), T, 0, stream>>>(
        cx1, nullptr, xd1, B, h1, w1, /*accumulate=*/0);
    iwt_kernel<<<blocks((long long)B * C_CH * H0 * W0, T), T, 0, stream>>>(
        cx0, xd1, out, B, h0, w0, /*accumulate=*/1);
}